// S4Layer_11192684773568
// MI455X (gfx1250) — compile-verified
//
#include <hip/hip_runtime.h>
#include <hip/hip_bf16.h>
#include <math.h>

#define LFULL  (1u << 20)   // L = 1048576 = 16^5
#define N2     (1u << 21)   // padded FFT length = 2*16^5
#define NMODES 64

typedef __attribute__((ext_vector_type(2))) float v2f;
typedef __attribute__((ext_vector_type(8))) float v8f;

// ---- gfx1250 async global->LDS helpers (ASYNCcnt path, CDNA5 ISA) -------
static __device__ __forceinline__ unsigned lds_off(const void* p) {
    // generic -> LDS addrspace -> byte offset from wave LDS base
    return (unsigned)(unsigned long long)(const __attribute__((address_space(3))) void*)p;
}
static __device__ __forceinline__ void async_b32_to_lds(unsigned dst, const float* g) {
    asm volatile("global_load_async_to_lds_b32 %0, %1, off"
                 :: "v"(dst), "v"(g) : "memory");
}
static __device__ __forceinline__ void wait_asynccnt0() {
    asm volatile("s_wait_asynccnt 0x0" ::: "memory");
}

// ---------------------------------------------------------------------------
// Stage 1: atRoots[l] = c(w) * (k00 - k01*k10/(1+k11)),  k_j = sum_n vs_j[n]/(g(w)-Lambda[n])
// Mode tables staged into LDS with async loads (broadcast tables, no VGPR bounce).
// ---------------------------------------------------------------------------
__global__ void __launch_bounds__(256)
atroots_kernel(const float* __restrict__ Bv,  const float* __restrict__ Ctr,
               const float* __restrict__ Cti, const float* __restrict__ logstep,
               const float* __restrict__ Lre, const float* __restrict__ Lim,
               const float* __restrict__ pre, const float* __restrict__ pim,
               const float* __restrict__ qre, const float* __restrict__ qim,
               float2* __restrict__ outA)
{
    __shared__ float sRaw[9][NMODES];                 // async-staged raw tables
    __shared__ float sLr[NMODES], sLi[NMODES];
    __shared__ float sVr[4][NMODES], sVi[4][NMODES];
    const int tid = threadIdx.x;

    if (tid < NMODES) {                               // waves 0-1: full EXEC
        async_b32_to_lds(lds_off(&sRaw[0][tid]), Bv  + tid);
        async_b32_to_lds(lds_off(&sRaw[1][tid]), Ctr + tid);
        async_b32_to_lds(lds_off(&sRaw[2][tid]), Cti + tid);
        async_b32_to_lds(lds_off(&sRaw[3][tid]), Lre + tid);
        async_b32_to_lds(lds_off(&sRaw[4][tid]), Lim + tid);
        async_b32_to_lds(lds_off(&sRaw[5][tid]), pre + tid);
        async_b32_to_lds(lds_off(&sRaw[6][tid]), pim + tid);
        async_b32_to_lds(lds_off(&sRaw[7][tid]), qre + tid);
        async_b32_to_lds(lds_off(&sRaw[8][tid]), qim + tid);
        wait_asynccnt0();
    }
    __syncthreads();

    if (tid < NMODES) {                               // derive vs_j, Lambda in LDS
        const float b0  = sRaw[0][tid];
        const float a0r = sRaw[1][tid], a0i = -sRaw[2][tid];   // conj(Ct)
        sLr[tid] = sRaw[3][tid];
        sLi[tid] = sRaw[4][tid];
        const float b1r = sRaw[5][tid], b1i = sRaw[6][tid];    // p
        const float a1r = sRaw[7][tid], a1i = -sRaw[8][tid];   // conj(q)
        sVr[0][tid] = a0r*b0;              sVi[0][tid] = a0i*b0;
        sVr[1][tid] = a0r*b1r - a0i*b1i;   sVi[1][tid] = a0r*b1i + a0i*b1r;
        sVr[2][tid] = a1r*b0;              sVi[2][tid] = a1i*b0;
        sVr[3][tid] = a1r*b1r - a1i*b1i;   sVi[3][tid] = a1r*b1i + a1i*b1r;
    }
    __syncthreads();

    const unsigned l = blockIdx.x * blockDim.x + tid;
    const float step = __expf(logstep[0]);
    const float th = 6.28318530717958647692f * ((float)l / (float)LFULL);
    const float cr = __cosf(th), ci = __sinf(th);     // Omega = e^{+i th}
    const float dr = 1.0f + cr, di = ci;              // 1+Omega
    const float inv = __frcp_rn(dr*dr + di*di);
    const float nr = 1.0f - cr, ni = -ci;             // 1-Omega
    const float rr = (nr*dr + ni*di) * inv;
    const float ri = (ni*dr - nr*di) * inv;
    const float gs = 2.0f / step;
    const float gr = gs*rr, gi = gs*ri;               // g(w)
    const float c2r =  2.0f*dr*inv;                   // c(w) = 2/(1+Omega)
    const float c2i = -2.0f*di*inv;

    float k0r=0.f,k0i=0.f,k1r=0.f,k1i=0.f,k2r=0.f,k2i=0.f,k3r=0.f,k3i=0.f;
    #pragma unroll 4
    for (int n = 0; n < NMODES; ++n) {
        const float ddr = gr - sLr[n], ddi = gi - sLi[n];
        const float iv  = __frcp_rn(ddr*ddr + ddi*ddi);
        const float crn =  ddr*iv, cin = -ddi*iv;     // 1/(g-Lambda)
        k0r += crn*sVr[0][n] - cin*sVi[0][n];  k0i += crn*sVi[0][n] + cin*sVr[0][n];
        k1r += crn*sVr[1][n] - cin*sVi[1][n];  k1i += crn*sVi[1][n] + cin*sVr[1][n];
        k2r += crn*sVr[2][n] - cin*sVi[2][n];  k2i += crn*sVi[2][n] + cin*sVr[2][n];
        k3r += crn*sVr[3][n] - cin*sVi[3][n];  k3i += crn*sVi[3][n] + cin*sVr[3][n];
    }
    const float er = 1.0f + k3r, ei = k3i;
    const float iv2 = __frcp_rn(er*er + ei*ei);
    const float tr = k1r*k2r - k1i*k2i, ti = k1r*k2i + k1i*k2r;
    const float sr = (tr*er + ti*ei)*iv2, si = (ti*er - tr*ei)*iv2;
    const float ar = k0r - sr, ai = k0i - si;
    outA[l] = make_float2(c2r*ar - c2i*ai, c2r*ai + c2i*ar);
}

// ---------------------------------------------------------------------------
// Stockham radix-16 DIF stage via V_WMMA_F32_16X16X4_F32, grid-stride tiles.
//   inputs  a_j  = x[t + j*(n/16)],  j = 0..15
//   DFT16 (16x16 matmul on the matrix pipe)
//   output  y[(t mod s) + 16*s*(t div s) + s*k1] * e^{-2pi i (t div s) k1 / ncur}
// A matrices are wave-uniform and hoisted out of the tile loop; output twiddles
// are generated incrementally (2 sincos/lane/tile). EXEC all-ones throughout.
// ---------------------------------------------------------------------------
__global__ void __launch_bounds__(256)
radix16_wmma_stage(const float2* __restrict__ x, float2* __restrict__ y,
                   int nlog2, int stage)
{
    const int lane  = threadIdx.x & 31;
    const unsigned nwaves = (gridDim.x * blockDim.x) >> 5;
    const unsigned wid    = (blockIdx.x * blockDim.x + threadIdx.x) >> 5;
    const unsigned n      = 1u << nlog2;
    const unsigned nb     = n >> 4;                   // total butterflies
    const unsigned ntiles = nb >> 4;

    // ---- A matrices: DFT16 W[m][j] = e^{-2pi i m j/16} (built once per wave)
    const int mrow = lane & 15;
    const int jadd = (lane >> 4) << 1;                // 0 (lanes 0-15) / 2 (16-31)
    v2f Are[4], Aim[4], Ain[4];
    #pragma unroll
    for (int c = 0; c < 4; ++c) {
        const int j0 = 4*c + jadd;
        const float a0 = -0.39269908169872415481f * (float)((mrow * j0)     & 15);
        const float a1 = -0.39269908169872415481f * (float)((mrow * (j0+1)) & 15);
        Are[c][0] = __cosf(a0);  Are[c][1] = __cosf(a1);
        Aim[c][0] = __sinf(a0);  Aim[c][1] = __sinf(a1);
        Ain[c][0] = -Aim[c][0];  Ain[c][1] = -Aim[c][1];
    }

    const int      slog2 = stage << 2;                // s = 16^stage
    const unsigned s     = 1u << slog2;
    // exact 1/ncur (power of two) -> no v_div sequence
    const float inv_ncur = __uint_as_float((unsigned)(127 - (nlog2 - slog2)) << 23);
    const int   k1base   = (lane >> 4) << 3;          // C layout rows v / v+8

    for (unsigned tile = wid; tile < ntiles; tile += nwaves) {
        const unsigned t0 = tile << 4;
        const unsigned tb = t0 + (unsigned)(lane & 15);

        // ---- B gather: lane<16 holds rows {4c,4c+1}, lane>=16 rows {4c+2,4c+3}
        v2f Bre[4], Bim[4];
        #pragma unroll
        for (int c = 0; c < 4; ++c) {
            const unsigned j0 = (unsigned)(4*c + jadd);
            const float2 e0 = x[tb + j0 * nb];
            const float2 e1 = x[tb + (j0 + 1u) * nb];
            Bre[c][0] = e0.x;  Bre[c][1] = e1.x;
            Bim[c][0] = e0.y;  Bim[c][1] = e1.y;
        }

        // ---- complex DFT16: Cre = Wre*Xre - Wim*Xim ; Cim = Wre*Xim + Wim*Xre
        v8f Cre = {}; v8f Cim = {};
        #pragma unroll
        for (int c = 0; c < 4; ++c) {
            Cre = __builtin_amdgcn_wmma_f32_16x16x4_f32(false, Are[c], false, Bre[c], (short)0, Cre, false, false);
            Cre = __builtin_amdgcn_wmma_f32_16x16x4_f32(false, Ain[c], false, Bim[c], (short)0, Cre, false, false);
            Cim = __builtin_amdgcn_wmma_f32_16x16x4_f32(false, Are[c], false, Bim[c], (short)0, Cim, false, false);
            Cim = __builtin_amdgcn_wmma_f32_16x16x4_f32(false, Aim[c], false, Bre[c], (short)0, Cim, false, false);
        }

        // ---- incremental output twiddle + coalesced scatter
        const unsigned q     = tb & (s - 1u);
        const unsigned p     = tb >> slog2;
        const unsigned obase = q + (p << (slog2 + 4));
        const float bang = -6.28318530717958647692f * ((float)(p * (unsigned)k1base) * inv_ncur);
        const float sang = -6.28318530717958647692f * ((float)p * inv_ncur);
        float wr = __cosf(bang), wi = __sinf(bang);
        const float swr = __cosf(sang), swi = __sinf(sang);
        #pragma unroll
        for (int v = 0; v < 8; ++v) {
            const float yr = Cre[v]*wr - Cim[v]*wi;
            const float yi = Cre[v]*wi + Cim[v]*wr;
            y[obase + ((unsigned)(k1base + v) << slog2)] = make_float2(yr, yi);
            const float nwr = wr*swr - wi*swi;        // w *= e^{-2pi i p/ncur}
            const float nwi = wr*swi + wi*swr;
            wr = nwr; wi = nwi;
        }
    }
}

// Final Stockham radix-2 stage for n = 2*16^5 (s = n/2, twiddle = 1).
__global__ void __launch_bounds__(256)
radix2_kernel(const float2* __restrict__ x, float2* __restrict__ y, unsigned half)
{
    const unsigned q = blockIdx.x * blockDim.x + threadIdx.x;
    const float2 a = x[q], b = x[q + half];
    y[q]        = make_float2(a.x + b.x, a.y + b.y);
    y[q + half] = make_float2(a.x - b.x, a.y - b.y);
}

// z[t] = u[t] + i*K[t] (t<L), 0 elsewhere;  K[t] = Re(FFT_L(atRoots)[t]) / L
__global__ void __launch_bounds__(256)
pack_z_kernel(const float* __restrict__ u, const float2* __restrict__ Kspec,
              float2* __restrict__ z)
{
    const unsigned t = blockIdx.x * blockDim.x + threadIdx.x;
    if (t < LFULL) {
        z[t] = make_float2(u[t], Kspec[t].x * (1.0f / (float)LFULL));
    } else {
        z[t] = make_float2(0.f, 0.f);
    }
}

// Split packed spectrum, multiply, and conjugate (inverse FFT via forward FFT).
__global__ void __launch_bounds__(256)
spectrum_kernel(const float2* __restrict__ Z, float2* __restrict__ Yc)
{
    const unsigned k = blockIdx.x * blockDim.x + threadIdx.x;
    const unsigned m = (N2 - k) & (N2 - 1u);
    const float2 zk = Z[k], zm = Z[m];
    const float2 U  = make_float2(0.5f*(zk.x + zm.x), 0.5f*(zk.y - zm.y));   // (Z[k]+conj(Z[m]))/2
    const float2 d  = make_float2(zk.x - zm.x, zk.y + zm.y);                 //  Z[k]-conj(Z[m])
    const float2 Kf = make_float2(0.5f*d.y, -0.5f*d.x);                      //  d/(2i)
    const float yr = U.x*Kf.x - U.y*Kf.y;
    const float yi = U.x*Kf.y + U.y*Kf.x;
    Yc[k] = make_float2(yr, -yi);                                            // conj(Y)
}

// out = Re(FFT(conj(Y)))[t]/(2L) + D*u[t]
__global__ void __launch_bounds__(256)
final_kernel(const float2* __restrict__ F, const float* __restrict__ u,
             const float* __restrict__ D, float* __restrict__ out)
{
    const unsigned t = blockIdx.x * blockDim.x + threadIdx.x;
    out[t] = F[t].x * (1.0f / (float)N2) + D[0] * u[t];
}

// ---------------------------------------------------------------------------
extern "C" void kernel_launch(void* const* d_in, const int* in_sizes, int n_in,
                              void* d_out, int out_size, void* d_ws, size_t ws_size,
                              hipStream_t stream)
{
    (void)in_sizes; (void)n_in; (void)out_size; (void)ws_size;
    const float* u   = (const float*)d_in[0];
    const float* B   = (const float*)d_in[1];
    const float* Ctr = (const float*)d_in[2];
    const float* Cti = (const float*)d_in[3];
    const float* D   = (const float*)d_in[4];
    const float* ls  = (const float*)d_in[5];
    const float* Lre = (const float*)d_in[6];
    const float* Lim = (const float*)d_in[7];
    const float* pre = (const float*)d_in[8];
    const float* pim = (const float*)d_in[9];
    const float* qre = (const float*)d_in[10];
    const float* qim = (const float*)d_in[11];
    float* out = (float*)d_out;

    float2* bufA = (float2*)d_ws;       // 2^21 complex
    float2* bufB = bufA + N2;           // 2^21 complex   (32 MB total, L2-resident)

    const dim3 blk(256);

    // 1) kernel generating function at the L roots of unity -> bufA
    atroots_kernel<<<LFULL/256, blk, 0, stream>>>(B, Ctr, Cti, ls, Lre, Lim,
                                                  pre, pim, qre, qim, bufA);

    // 2) FFT_L(atRoots): 5 radix-16 WMMA stages (4 tiles/wave), result bufB
    {
        const unsigned blocks = 128;                 // 1024 waves, 4096 tiles
        float2 *src = bufA, *dst = bufB;
        for (int s = 0; s < 5; ++s) {
            radix16_wmma_stage<<<blocks, blk, 0, stream>>>(src, dst, 20, s);
            float2* t = src; src = dst; dst = t;
        }
    }

    // 3) z = u + i*K, zero-padded to 2L -> bufA
    pack_z_kernel<<<N2/256, blk, 0, stream>>>(u, bufB, bufA);

    // 4) FFT_{2L}(z): 5 radix-16 stages + radix-2 (result bufA)
    {
        const unsigned blocks = 256;                 // 2048 waves, 8192 tiles
        float2 *src = bufA, *dst = bufB;
        for (int s = 0; s < 5; ++s) {
            radix16_wmma_stage<<<blocks, blk, 0, stream>>>(src, dst, 21, s);
            float2* t = src; src = dst; dst = t;
        }
        radix2_kernel<<<(N2/2)/256, blk, 0, stream>>>(src, dst, N2/2);  // bufB -> bufA
    }

    // 5) split spectra, Y = U*Kf, store conj(Y) -> bufB
    spectrum_kernel<<<N2/256, blk, 0, stream>>>(bufA, bufB);

    // 6) inverse FFT via forward FFT of conj(Y): result bufB
    {
        const unsigned blocks = 256;
        float2 *src = bufB, *dst = bufA;
        for (int s = 0; s < 5; ++s) {
            radix16_wmma_stage<<<blocks, blk, 0, stream>>>(src, dst, 21, s);
            float2* t = src; src = dst; dst = t;
        }
        radix2_kernel<<<(N2/2)/256, blk, 0, stream>>>(src, dst, N2/2);  // bufA -> bufB
    }

    // 7) y = Re(.)/(2L) + D*u
    final_kernel<<<LFULL/256, blk, 0, stream>>>(bufB, u, D, out);
}